// DenseCapsule_31954556682285
// MI455X (gfx1250) — compile-verified
//
#include <hip/hip_runtime.h>
#include <math.h>

typedef __attribute__((ext_vector_type(2))) float v2f;
typedef __attribute__((ext_vector_type(4))) float v4f;
typedef __attribute__((ext_vector_type(8))) float v8f;

#define BATCH 64
#define IN_N  2048
#define IN_D  8
#define OUT_N 32
#define OUT_D 16
#define KT    8    // in-capsules (k) per block
#define BT    16   // batch elements per block (WMMA N dimension)

// ---------------------------------------------------------------------------
// Per-routing-pass kernel.
// Block = (b-tile of 16, k-tile of 8). 32 waves; wave w handles out-capsule j=w.
// Recomputes x_hat[p,b] tiles with V_WMMA_F32_16X16X4_F32 (2 chained per (j,k)),
// computes logits = O . x_hat, softmax over j in LDS, then atomically
// accumulates s[b,j,p] += c * x_hat.
// ---------------------------------------------------------------------------
__global__ __launch_bounds__(1024) void caps_pass_kernel(
    const float* __restrict__ x,      // [B, IN_N, IN_D]
    const float* __restrict__ weight, // [OUT_N, IN_N, OUT_D, IN_D]
    const float* __restrict__ Oacc,   // [B, OUT_N, OUT_D]  (sum of prior outputs)
    float* __restrict__ s,            // [B, OUT_N, OUT_D]  (pass partial, pre-zeroed)
    int pass)
{
    __shared__ float lds_c[KT * BT * OUT_N];   // [k][b][j]: logits -> coupling c

    const int lane = threadIdx.x & 31;
    const int j    = threadIdx.x >> 5;   // wave id == out capsule index
    const int bb   = lane & 15;          // local batch idx (B-matrix N / A-matrix M row)
    const int hi   = lane >> 4;          // lane half selects K/M sub-range

    const int b0 = (blockIdx.x & 3) * BT;
    const int k0 = (blockIdx.x >> 2) * KT;

    // ---- B fragments: x[q, b] for each k in the tile -------------------------
    v2f xb0[KT], xb1[KT];
#pragma unroll
    for (int k = 0; k < KT; ++k) {
        const float* xp = x + ((b0 + bb) * IN_N + (k0 + k)) * IN_D + 2 * hi;
        xb0[k] = *(const v2f*)xp;        // q = 2*hi, 2*hi+1
        xb1[k] = *(const v2f*)(xp + 4);  // q = 4+2*hi, 4+2*hi+1
    }

    // ---- x_hat tiles via f32 WMMA: D[p,b] = W[p,q] * x[q,b] ------------------
    v8f xh[KT];
#pragma unroll
    for (int k = 0; k < KT; ++k) {
        const float* wp = weight + ((j * IN_N + (k0 + k)) * OUT_D + bb) * IN_D + 2 * hi;
        v2f a0 = *(const v2f*)wp;        // A[p, q=2*hi .. 2*hi+1]
        v2f a1 = *(const v2f*)(wp + 4);  // A[p, q=4+2*hi ..]
        v8f d = {};
        d = __builtin_amdgcn_wmma_f32_16x16x4_f32(false, a0, false, xb0[k],
                                                  (short)0, d, false, false);
        d = __builtin_amdgcn_wmma_f32_16x16x4_f32(false, a1, false, xb1[k],
                                                  (short)0, d, false, false);
        xh[k] = d;   // lane L holds x_hat[p = 8*hi + r, b = L&15], r = 0..7
    }

    if (pass > 0) {
        // O fragment matches the D layout: lane L needs O[b, j, 8*hi + 0..7]
        const float* op = Oacc + ((b0 + bb) * OUT_N + j) * OUT_D + 8 * hi;
        v4f o0 = *(const v4f*)op;
        v4f o1 = *(const v4f*)(op + 4);

#pragma unroll
        for (int k = 0; k < KT; ++k) {
            float t = 0.f;
#pragma unroll
            for (int r = 0; r < 4; ++r) t += o0[r] * xh[k][r];
#pragma unroll
            for (int r = 0; r < 4; ++r) t += o1[r] * xh[k][4 + r];
            // combine the two p-halves (lanes L and L^16), wave32 shuffle
            t += __shfl_xor(t, 16, 32);
            if (hi == 0)
                lds_c[(k * BT + bb) * OUT_N + j] = t;   // logit[b, j, k]
        }
        __syncthreads();

        // softmax over j (axis=1) for each (k, b) pair: 128 rows of 32
        if (threadIdx.x < KT * BT) {
            float* row = &lds_c[threadIdx.x * OUT_N];
            float m = row[0];
#pragma unroll
            for (int jj = 1; jj < OUT_N; ++jj) m = fmaxf(m, row[jj]);
            float sum = 0.f;
#pragma unroll
            for (int jj = 0; jj < OUT_N; ++jj) {
                float e = __expf(row[jj] - m);
                row[jj] = e;
                sum += e;
            }
            float inv = 1.f / sum;
#pragma unroll
            for (int jj = 0; jj < OUT_N; ++jj) row[jj] *= inv;
        }
        __syncthreads();
    }

    // ---- s[b,j,p] += sum_k c[b,j,k] * x_hat[b,j,k,p] -------------------------
    v8f acc = {};
#pragma unroll
    for (int k = 0; k < KT; ++k) {
        float c = (pass == 0) ? (1.0f / OUT_N)
                              : lds_c[(k * BT + bb) * OUT_N + j];
#pragma unroll
        for (int r = 0; r < 8; ++r) acc[r] += c * xh[k][r];
    }

    float* sp = s + ((b0 + bb) * OUT_N + j) * OUT_D + 8 * hi;
#pragma unroll
    for (int r = 0; r < 8; ++r) atomicAdd(sp + r, acc[r]);
}

// ---------------------------------------------------------------------------
// squash(s) -> either accumulate into O (intermediate passes) or write d_out.
// ---------------------------------------------------------------------------
__global__ void caps_squash_kernel(const float* __restrict__ s,
                                   float* __restrict__ Oacc,
                                   float* __restrict__ outp,
                                   int final_pass)
{
    int idx = blockIdx.x * blockDim.x + threadIdx.x;   // (b*OUT_N + j)
    if (idx >= BATCH * OUT_N) return;
    const float* sp = s + idx * OUT_D;
    float v[OUT_D];
    float n2 = 0.f;
#pragma unroll
    for (int p = 0; p < OUT_D; ++p) { v[p] = sp[p]; n2 += v[p] * v[p]; }
    float norm  = sqrtf(n2);
    float scale = norm / (1.0f + n2);   // ||s|| / (lmd + ||s||^2), lmd = 1
    if (final_pass) {
#pragma unroll
        for (int p = 0; p < OUT_D; ++p) outp[idx * OUT_D + p] = v[p] * scale;
    } else {
#pragma unroll
        for (int p = 0; p < OUT_D; ++p) Oacc[idx * OUT_D + p] += v[p] * scale;
    }
}

__global__ void caps_zero_kernel(float* __restrict__ p, int n)
{
    int i = blockIdx.x * blockDim.x + threadIdx.x;
    if (i < n) p[i] = 0.f;
}

// ---------------------------------------------------------------------------
extern "C" void kernel_launch(void* const* d_in, const int* in_sizes, int n_in,
                              void* d_out, int out_size, void* d_ws, size_t ws_size,
                              hipStream_t stream)
{
    const float* x = (const float*)d_in[0];   // [64, 2048, 8]
    const float* w = (const float*)d_in[1];   // [32, 2048, 16, 8]
    float* outp = (float*)d_out;              // [64, 32, 16]

    const int SN = BATCH * OUT_N * OUT_D;     // 32768 floats
    float* s = (float*)d_ws;                  // pass partial sums
    float* O = s + SN;                        // accumulated outputs (out0, out0+out1)

    const dim3 passGrid((IN_N / KT) * (BATCH / BT));   // 256 * 4 = 1024 blocks
    const dim3 passBlock(1024);                        // 32 waves (one per out-capsule)
    const dim3 sqGrid((BATCH * OUT_N + 255) / 256);
    const dim3 zGrid((SN + 255) / 256);

    // zero s and O (workspace is poisoned by the harness; must re-zero each call)
    caps_zero_kernel<<<dim3((2 * SN + 255) / 256), 256, 0, stream>>>(s, 2 * SN);

    // pass 0: c uniform (softmax of zeros), O = squash(s0)
    caps_pass_kernel<<<passGrid, passBlock, 0, stream>>>(x, w, O, s, 0);
    caps_squash_kernel<<<sqGrid, 256, 0, stream>>>(s, O, outp, 0);

    // pass 1: c = softmax_j(O . x_hat), O += squash(s1)
    caps_zero_kernel<<<zGrid, 256, 0, stream>>>(s, SN);
    caps_pass_kernel<<<passGrid, passBlock, 0, stream>>>(x, w, O, s, 1);
    caps_squash_kernel<<<sqGrid, 256, 0, stream>>>(s, O, outp, 0);

    // pass 2: final coupling, d_out = squash(s2)
    caps_zero_kernel<<<zGrid, 256, 0, stream>>>(s, SN);
    caps_pass_kernel<<<passGrid, passBlock, 0, stream>>>(x, w, O, s, 2);
    caps_squash_kernel<<<sqGrid, 256, 0, stream>>>(s, O, outp, 1);
}